// Lagrangian_motion_estimate_net_38104949850358
// MI455X (gfx1250) — compile-verified
//
#include <hip/hip_runtime.h>
#include <hip/hip_bf16.h>
#include <math.h>

// ---------------------------------------------------------------------------
// Lagrangian motion-estimation net for MI455X (gfx1250, wave32).
// Convs = implicit GEMM on V_WMMA_F32_16X16X4_F32 (fp32-exact, matrix pipe).
// Each wave: 16 Cout x 64 px (4 N-tiles sharing each A fragment).
// Upsample+concat fused into conv input fetch. Warps are elementwise gathers.
// ---------------------------------------------------------------------------

typedef float v2f __attribute__((ext_vector_type(2)));
typedef float v8f __attribute__((ext_vector_type(8)));

#define NB 32
#define HH 192
#define WW 192
#define HWFULL (HH * WW)

// ============================ WMMA conv kernel =============================
// A (weights) 16x4 f32 frag: lanes 0-15 M=0..15 hold K = kk+{0,1};
//                            lanes 16-31 hold K = kk+{2,3}.   (ISA 7.12.2)
// B (im2col)  4x16 f32 frag: mirrored — lane&15 = N, half-wave selects K pair.
// C/D 16x16 f32: VGPR r -> M = r (+8 for lanes 16-31), N = lane&15.
// Input = [A (Ca ch, optionally nearest-2x-upsampled)] ++ [B (Cb ch, full res)]
template <int KS>
__global__ __launch_bounds__(32) void k_conv_wmma(
    const float* __restrict__ A, int Ca, int upA,
    const float* __restrict__ Bp, int Cb,
    const float* __restrict__ Wt, const float* __restrict__ bias,
    float* __restrict__ out, int Cout, int Hout, int Wout, int Hin, int Win,
    int leaky)
{
  constexpr int KK  = KS * KS;          // 9 or 16
  constexpr int KPC = (KK + 3) & ~3;    // per-channel K padded to mult. of 4
  constexpr int NT  = 4;                // spatial 16-px tiles per wave
  constexpr int STR = (KS == 4) ? 2 : 1;

  const int lane   = threadIdx.x;       // 0..31 (wave32)
  const int hiHalf = lane >> 4;         // 0 -> K {0,1}; 1 -> K {2,3}
  const int nIdx   = lane & 15;
  const int ocBase = blockIdx.y * 16;
  const int n      = blockIdx.z;
  const int HWo    = Hout * Wout;
  const int Cin    = Ca + Cb;
  const int Ktot   = Cin * KK;
  const int pBase  = blockIdx.x * (16 * NT);

  // hoist per-tile spatial decomposition (NT runtime divisions total)
  int oy[NT], ox[NT];
  bool live[NT];
#pragma unroll
  for (int t = 0; t < NT; ++t) {
    const int pos = pBase + 16 * t + nIdx;
    live[t] = (pos < HWo);
    const int pp = live[t] ? pos : 0;
    oy[t] = pp / Wout;
    ox[t] = pp - oy[t] * Wout;
  }

  const int  ocA     = ocBase + nIdx;        // A-frag row for this lane
  const bool ocValid = (ocA < Cout);

  v8f acc[NT] = {};

  for (int ci = 0; ci < Cin; ++ci) {
    // uniform (SALU) per-channel input plane selection
    const bool fromA = (ci < Ca);
    const int  up    = fromA ? upA : 0;
    const int  pH    = up ? (Hin >> 1) : Hin;
    const int  pW    = up ? (Win >> 1) : Win;
    const float* plane =
        fromA ? A + ((size_t)n * Ca + ci) * ((size_t)pH * pW)
              : Bp + ((size_t)n * Cb + (ci - Ca)) * ((size_t)Hin * Win);
    const size_t wbase = (size_t)ocA * Ktot + (size_t)ci * KK;

#pragma unroll
    for (int chunk = 0; chunk < KPC / 4; ++chunk) {
      const int r0 = chunk * 4 + 2 * hiHalf;
      // ---- A fragment (weights, shared by all NT tiles) ----
      v2f afrag = {};
      if (ocValid) {
        if (r0 < KK)     afrag.x = Wt[wbase + r0];
        if (r0 + 1 < KK) afrag.y = Wt[wbase + r0 + 1];
      }
      // ---- NT B fragments + WMMAs ----
#pragma unroll
      for (int t = 0; t < NT; ++t) {
        v2f bfrag = {};
#pragma unroll
        for (int tt = 0; tt < 2; ++tt) {
          const int r = r0 + tt;
          float v = 0.0f;
          if (r < KK && live[t]) {
            const int ky = r / KS;          // constant-divisor, unrolled
            const int kx = r - ky * KS;
            int iy = oy[t] * STR + ky - 1;  // pad = 1
            int ix = ox[t] * STR + kx - 1;
            if (iy >= 0 && iy < Hin && ix >= 0 && ix < Win) {
              if (up) { iy >>= 1; ix >>= 1; }
              v = plane[iy * pW + ix];
            }
          }
          if (tt == 0) bfrag.x = v; else bfrag.y = v;
        }
        acc[t] = __builtin_amdgcn_wmma_f32_16x16x4_f32(
            false, afrag, false, bfrag, (short)0, acc[t], false, false);
      }
    }
  }

  // ---- epilogue: bias + leaky ReLU + scatter D tiles ----
#pragma unroll
  for (int t = 0; t < NT; ++t) {
    if (!live[t]) continue;
    const int pos = pBase + 16 * t + nIdx;
#pragma unroll
    for (int r = 0; r < 8; ++r) {
      const int m = r + hiHalf * 8;
      const int o = ocBase + m;
      if (o < Cout) {
        float v = acc[t][r] + bias[o];
        if (leaky) v = (v >= 0.0f) ? v : 0.2f * v;
        out[((size_t)n * Cout + o) * HWo + pos] = v;
      }
    }
  }
}

// ============================ elementwise kernels ==========================
__global__ void k_pack_x(const float* __restrict__ src,
                         const float* __restrict__ tgt,
                         float* __restrict__ xcat)
{
  int idx = blockIdx.x * blockDim.x + threadIdx.x;
  const int total = NB * 2 * HWFULL;
  if (idx >= total) return;
  int n = idx / (2 * HWFULL);
  int rem = idx - n * 2 * HWFULL;
  int c = rem / HWFULL;
  int p = rem - c * HWFULL;
  xcat[idx] = (c == 0 ? src : tgt)[(size_t)n * HWFULL + p];
}

__global__ void k_sample_flow(const float* __restrict__ mean,
                              const float* __restrict__ ls,
                              const float* __restrict__ noise,
                              float* __restrict__ fpos,
                              float* __restrict__ fneg, float scale)
{
  int idx = blockIdx.x * blockDim.x + threadIdx.x;
  const int total = NB * 2 * HWFULL;
  if (idx >= total) return;
  float f = mean[idx] + __expf(0.5f * ls[idx]) * noise[idx];
  fpos[idx] = f * scale;
  fneg[idx] = -f * scale;
}

__device__ __forceinline__ float bilin(const float* __restrict__ img, float sy,
                                       float sx, int H, int W)
{
  const float y0f = floorf(sy), x0f = floorf(sx);
  const float ty = sy - y0f, tx = sx - x0f;
  const int y0 = (int)y0f, x0 = (int)x0f;
  float acc = 0.0f;
#pragma unroll
  for (int dy = 0; dy < 2; ++dy) {
#pragma unroll
    for (int dx = 0; dx < 2; ++dx) {
      // validity on unclamped coordinate (matches reference zero padding)
      const float yi = y0f + (float)dy, xi = x0f + (float)dx;
      const bool valid = (yi >= 0.0f) && (yi <= (float)(H - 1)) &&
                         (xi >= 0.0f) && (xi <= (float)(W - 1));
      const int yc = min(max(y0 + dy, 0), H - 1);
      const int xc = min(max(x0 + dx, 0), W - 1);
      const float w = (dy ? ty : 1.0f - ty) * (dx ? tx : 1.0f - tx);
      acc += valid ? img[(size_t)yc * W + xc] * w : 0.0f;
    }
  }
  return acc;
}

// out[n,ch] = flow[n,ch] + bilinear(srcF[n,ch], y+flow0, x+flow1)   (C == 2)
__global__ void k_warp_compose(float* __restrict__ out,
                               const float* __restrict__ srcF,
                               const float* __restrict__ flow, int NN)
{
  int idx = blockIdx.x * blockDim.x + threadIdx.x;
  const int total = NN * HWFULL;
  if (idx >= total) return;
  int n = idx / HWFULL;
  int p = idx - n * HWFULL;
  int y = p / WW, x = p - y * WW;
  const float* f0 = flow + (size_t)(2 * n) * HWFULL;
  const float fy = f0[p], fx = f0[HWFULL + p];
  const float sy = (float)y + fy, sx = (float)x + fx;
  const float* s0 = srcF + (size_t)(2 * n) * HWFULL;
  out[(size_t)(2 * n) * HWFULL + p]          = fy + bilin(s0, sy, sx, HH, WW);
  out[(size_t)(2 * n) * HWFULL + HWFULL + p] = fx + bilin(s0 + HWFULL, sy, sx, HH, WW);
}

// out[n,0] = bilinear(img[broadcast ? 0 : n], y+flow[n,0], x+flow[n,1])
__global__ void k_warp_img(float* __restrict__ out, const float* __restrict__ img,
                           const float* __restrict__ flow, int broadcast)
{
  int idx = blockIdx.x * blockDim.x + threadIdx.x;
  const int total = NB * HWFULL;
  if (idx >= total) return;
  int n = idx / HWFULL;
  int p = idx - n * HWFULL;
  int y = p / WW, x = p - y * WW;
  const float* f0 = flow + (size_t)(2 * n) * HWFULL;
  const float sy = (float)y + f0[p];
  const float sx = (float)x + f0[HWFULL + p];
  const float* im = broadcast ? img : img + (size_t)n * HWFULL;
  out[idx] = bilin(im, sy, sx, HH, WW);
}

// ================================ host side ================================
static void launch_conv(hipStream_t stream, int KS, const float* A, int Ca,
                        int upA, const float* Bp, int Cb, const float* Wt,
                        const float* bias, float* out, int Cout, int Ho, int Wo,
                        int leaky)
{
  const int Hin = (KS == 4) ? Ho * 2 : Ho;
  const int Win = (KS == 4) ? Wo * 2 : Wo;
  dim3 g((Ho * Wo + 63) / 64, (Cout + 15) / 16, NB);  // 64 px per wave
  if (KS == 4)
    k_conv_wmma<4><<<g, 32, 0, stream>>>(A, Ca, upA, Bp, Cb, Wt, bias, out,
                                         Cout, Ho, Wo, Hin, Win, leaky);
  else
    k_conv_wmma<3><<<g, 32, 0, stream>>>(A, Ca, upA, Bp, Cb, Wt, bias, out,
                                         Cout, Ho, Wo, Hin, Win, leaky);
}

extern "C" void kernel_launch(void* const* d_in, const int* in_sizes, int n_in,
                              void* d_out, int out_size, void* d_ws,
                              size_t ws_size, hipStream_t stream)
{
  (void)in_sizes; (void)n_in; (void)out_size; (void)ws_size;

  // ---- inputs (setup_inputs order; params pytree with sorted dict keys) ----
  const float* src   = (const float*)d_in[0];
  const float* tgt   = (const float*)d_in[1];
  const float* noise = (const float*)d_in[2];
  int pi = 3;
  const float *decW[7], *decB[7];
  for (int i = 0; i < 7; ++i) { decW[i] = (const float*)d_in[pi++]; decB[i] = (const float*)d_in[pi++]; }
  const float *encW[4], *encB[4];
  for (int i = 0; i < 4; ++i) { encW[i] = (const float*)d_in[pi++]; encB[i] = (const float*)d_in[pi++]; }
  const float* lsW   = (const float*)d_in[pi++];
  const float* lsB   = (const float*)d_in[pi++];
  const float* meanW = (const float*)d_in[pi++];
  const float* meanB = (const float*)d_in[pi++];

  // ---- output layout (reference return order, flat concat) ----
  float* out    = (float*)d_out;
  float* y_src  = out;
  float* y_tgt  = y_src + (size_t)NB * HWFULL;
  float* lag_y  = y_tgt + (size_t)NB * HWFULL;
  float* inf_f  = lag_y + (size_t)NB * HWFULL;
  float* neg_f  = inf_f + (size_t)NB * 2 * HWFULL;
  float* lag_f  = neg_f + (size_t)NB * 2 * HWFULL;
  float* f_mean = lag_f + (size_t)NB * 2 * HWFULL;
  float* f_ls   = f_mean + (size_t)NB * 2 * HWFULL;

  // ---- workspace carve-up (~317 MB fp32 intermediates) ----
  float* ws = (float*)d_ws;
  size_t o = 0;
  auto take = [&](size_t nf) { float* p = ws + o; o += nf; return p; };
  float* xcat = take((size_t)NB * 2 * HWFULL);
  float* e1   = take((size_t)NB * 16 * 96 * 96);
  float* e2   = take((size_t)NB * 32 * 48 * 48);
  float* e3   = take((size_t)NB * 32 * 24 * 24);
  float* e4   = take((size_t)NB * 32 * 12 * 12);
  float* t0   = take((size_t)NB * 32 * 12 * 12);   // dec0 out
  float* t1   = take((size_t)NB * 32 * 24 * 24);   // dec1 out
  float* t2   = take((size_t)NB * 32 * 48 * 48);   // dec2 out
  float* t3   = take((size_t)NB * 32 * 96 * 96);   // dec3 out
  float* t4   = take((size_t)NB * 32 * 96 * 96);   // dec4 out
  float* t5   = take((size_t)NB * 16 * HWFULL);    // dec5 out
  float* t6   = take((size_t)NB * 16 * HWFULL);    // dec6 out (unet feat)
  float* fpA  = take((size_t)NB * 2 * HWFULL);     // diffeo ping-pong (+flow)
  float* fpB  = take((size_t)NB * 2 * HWFULL);
  float* fnA  = take((size_t)NB * 2 * HWFULL);     // diffeo ping-pong (-flow)
  float* fnB  = take((size_t)NB * 2 * HWFULL);

  const int EB = 256;

  // ---- x = concat([src, tgt], axis=1) ----
  k_pack_x<<<(NB * 2 * HWFULL + EB - 1) / EB, EB, 0, stream>>>(src, tgt, xcat);

  // ---- UNet encoder (k=4, s=2, p=1, leaky) ----
  launch_conv(stream, 4, xcat, 2, 0, nullptr, 0, encW[0], encB[0], e1, 16, 96, 96, 1);
  launch_conv(stream, 4, e1, 16, 0, nullptr, 0, encW[1], encB[1], e2, 32, 48, 48, 1);
  launch_conv(stream, 4, e2, 32, 0, nullptr, 0, encW[2], encB[2], e3, 32, 24, 24, 1);
  launch_conv(stream, 4, e3, 32, 0, nullptr, 0, encW[3], encB[3], e4, 32, 12, 12, 1);

  // ---- UNet decoder (k=3, s=1, p=1); upsample+concat fused into fetch ----
  launch_conv(stream, 3, e4, 32, 0, nullptr, 0, decW[0], decB[0], t0, 32, 12, 12, 1);
  launch_conv(stream, 3, t0, 32, 1, e3, 32,    decW[1], decB[1], t1, 32, 24, 24, 1);
  launch_conv(stream, 3, t1, 32, 1, e2, 32,    decW[2], decB[2], t2, 32, 48, 48, 1);
  launch_conv(stream, 3, t2, 32, 1, e1, 16,    decW[3], decB[3], t3, 32, 96, 96, 1);
  launch_conv(stream, 3, t3, 32, 0, nullptr, 0, decW[4], decB[4], t4, 32, 96, 96, 1);
  launch_conv(stream, 3, t4, 32, 1, xcat, 2,   decW[5], decB[5], t5, 16, HH, WW, 1);
  launch_conv(stream, 3, t5, 16, 0, nullptr, 0, decW[6], decB[6], t6, 16, HH, WW, 1);

  // ---- flow heads (no activation), written straight to d_out ----
  launch_conv(stream, 3, t6, 16, 0, nullptr, 0, meanW, meanB, f_mean, 2, HH, WW, 0);
  launch_conv(stream, 3, t6, 16, 0, nullptr, 0, lsW,   lsB,   f_ls,   2, HH, WW, 0);

  // ---- reparameterized sample; seed both integration directions ----
  k_sample_flow<<<(NB * 2 * HWFULL + EB - 1) / EB, EB, 0, stream>>>(
      f_mean, f_ls, noise, fpA, fnA, 1.0f / 128.0f);  // 2^-INT_STEPS

  // ---- scaling-and-squaring: flow <- flow + warp(flow, flow), 7 steps ----
  const int WG = (NB * HWFULL + EB - 1) / EB;
  {
    float* cur = fpA;
    for (int s = 0; s < 7; ++s) {
      float* dst = (s == 6) ? inf_f : ((s & 1) ? fpA : fpB);
      k_warp_compose<<<WG, EB, 0, stream>>>(dst, cur, cur, NB);
      cur = dst;
    }
    cur = fnA;
    for (int s = 0; s < 7; ++s) {
      float* dst = (s == 6) ? neg_f : ((s & 1) ? fnA : fnB);
      k_warp_compose<<<WG, EB, 0, stream>>>(dst, cur, cur, NB);
      cur = dst;
    }
  }

  // ---- warped images ----
  k_warp_img<<<WG, EB, 0, stream>>>(y_src, src, inf_f, 0);
  k_warp_img<<<WG, EB, 0, stream>>>(y_tgt, tgt, neg_f, 0);

  // ---- Lagrangian composition: v1k = v1(k-1) + warp(u_k, v1(k-1)) ----
  hipMemcpyAsync(lag_f, inf_f, (size_t)2 * HWFULL * sizeof(float),
                 hipMemcpyDeviceToDevice, stream);
  const int WG1 = (HWFULL + EB - 1) / EB;
  for (int k = 1; k < NB; ++k) {
    k_warp_compose<<<WG1, EB, 0, stream>>>(lag_f + (size_t)k * 2 * HWFULL,
                                           inf_f + (size_t)k * 2 * HWFULL,
                                           lag_f + (size_t)(k - 1) * 2 * HWFULL,
                                           1);
  }
  // lag_y_src = warp(broadcast(src[0]), lag_flow)
  k_warp_img<<<WG, EB, 0, stream>>>(lag_y, src, lag_f, 1);
}